// StackedGCN_16819091931637
// MI455X (gfx1250) — compile-verified
//
#include <hip/hip_runtime.h>
#include <math.h>

// ---------------------------------------------------------------------------
// StackedGCN for MI455X (gfx1250, wave32).
//  - Dense transforms via V_WMMA_F32_16X16X4_F32 (fp32-exact matrix pipe),
//    fully unrolled K-loop (templated dims), 2 independent accumulator
//    chains per wave to hide WMMA latency, ReLU fused at compile time.
//  - Edge aggregation via coalesced float4 gathers + global_atomic_add_f32.
//  - Final log_softmax: one wave per 32-wide row, shfl_xor reductions.
// ---------------------------------------------------------------------------

typedef float v2f __attribute__((ext_vector_type(2)));
typedef float v8f __attribute__((ext_vector_type(8)));

__device__ __forceinline__ v8f wmma_f32_16x16x4(v2f a, v2f b, v8f c) {
    return __builtin_amdgcn_wmma_f32_16x16x4_f32(
        /*neg_a=*/false, a, /*neg_b=*/false, b,
        /*c_mod=*/(short)0, c, /*reuse_a=*/false, /*reuse_b=*/false);
}

// ---------------- degree / normalization ----------------

__global__ void gcn_deg_init(float* deg, int n) {
    int i = blockIdx.x * blockDim.x + threadIdx.x;
    if (i < n) deg[i] = 1.0f;   // self loop contributes 1
}

__global__ void gcn_deg_count(const int* __restrict__ dst, float* deg, int e) {
    int i = blockIdx.x * blockDim.x + threadIdx.x;
    if (i < e) atomicAdd(&deg[dst[i]], 1.0f);
}

__global__ void gcn_deg_rsqrt(float* deg, int n) {
    int i = blockIdx.x * blockDim.x + threadIdx.x;
    if (i < n) deg[i] = rsqrtf(deg[i]);   // deg >= 1 always (self loops)
}

// ---------------- dense transform: H = act(X) @ W (templated fast path) ----
// One wave computes a 16x32 strip of H: two adjacent 16x16 tiles sharing the
// A fragment -> two independent WMMA accumulator chains.
template<int IND, int OUTD, bool RELU>
__global__ void gcn_gemm_wmma_t(const float* __restrict__ X,
                                const float* __restrict__ W,
                                float* __restrict__ H, int n) {
    constexpr int STRIPS = OUTD / 32;     // 16x32 strips per row-block
    int wave  = (blockIdx.x * blockDim.x + threadIdx.x) >> 5;
    int lane  = threadIdx.x & 31;
    int mTile = wave / STRIPS;
    int pTile = wave - mTile * STRIPS;
    if (mTile * 16 >= n) return;

    int half = lane >> 4;                 // 0: lanes 0-15, 1: lanes 16-31
    int l16  = lane & 15;
    int row  = mTile * 16 + l16;          // A: M = lane%16
    int col0 = pTile * 32 + l16;          // first 16-col tile
    int col1 = col0 + 16;                 // second 16-col tile

    const float* xp = X + (size_t)row * IND;
    v8f acc0 = {}, acc1 = {};
    #pragma unroll
    for (int kb = 0; kb < IND; kb += 4) {
        int k = kb + 2 * half;            // this lane-half supplies K = k, k+1
        v2f a;
        a.x = xp[k + 0];
        a.y = xp[k + 1];
        if (RELU) { a.x = fmaxf(a.x, 0.0f); a.y = fmaxf(a.y, 0.0f); }
        v2f b0, b1;
        b0.x = W[(k + 0) * OUTD + col0];
        b0.y = W[(k + 1) * OUTD + col0];
        b1.x = W[(k + 0) * OUTD + col1];
        b1.y = W[(k + 1) * OUTD + col1];
        acc0 = wmma_f32_16x16x4(a, b0, acc0);
        acc1 = wmma_f32_16x16x4(a, b1, acc1);
    }

    // D layout: VGPR r -> M = r + 8*half, N = lane%16
    int mBase = mTile * 16 + 8 * half;
    #pragma unroll
    for (int r = 0; r < 8; ++r) {
        H[(size_t)(mBase + r) * OUTD + col0] = acc0[r];
        H[(size_t)(mBase + r) * OUTD + col1] = acc1[r];
    }
}

// ---------------- generic fallback (runtime dims) ----------------
__global__ void gcn_gemm_wmma_g(const float* __restrict__ X,
                                const float* __restrict__ W,
                                float* __restrict__ H,
                                int n, int inD, int outD, int relu) {
    int wave  = (blockIdx.x * blockDim.x + threadIdx.x) >> 5;
    int lane  = threadIdx.x & 31;
    int tilesN = outD >> 4;
    int mTile = wave / tilesN;
    int nTile = wave - mTile * tilesN;
    if (mTile * 16 >= n) return;
    int half = lane >> 4, l16 = lane & 15;
    int row = mTile * 16 + l16;
    int col = nTile * 16 + l16;
    v8f acc = {};
    for (int kb = 0; kb < inD; kb += 4) {
        int k = kb + 2 * half;
        v2f a; a.x = X[row * inD + k]; a.y = X[row * inD + k + 1];
        if (relu) { a.x = fmaxf(a.x, 0.0f); a.y = fmaxf(a.y, 0.0f); }
        v2f b; b.x = W[(k + 0) * outD + col]; b.y = W[(k + 1) * outD + col];
        acc = wmma_f32_16x16x4(a, b, acc);
    }
    int mBase = mTile * 16 + 8 * half;
    #pragma unroll
    for (int r = 0; r < 8; ++r)
        H[(mBase + r) * outD + col] = acc[r];
}

// ---------------- aggregation init: out = b + dinv[i]^2 * h[i] --------------
__global__ void gcn_agg_init(const float* __restrict__ H,
                             const float* __restrict__ dinv,
                             const float* __restrict__ bias,
                             float* __restrict__ out, int n, int D) {
    int tid = blockIdx.x * blockDim.x + threadIdx.x;
    int vecPerRow = D >> 2;
    int i  = tid / vecPerRow;
    int d4 = (tid - i * vecPerRow) * 4;
    if (i >= n) return;
    float di = dinv[i];
    float w  = di * di;
    float4 h = *reinterpret_cast<const float4*>(H + (size_t)i * D + d4);
    float4 o;
    o.x = bias[d4 + 0] + w * h.x;
    o.y = bias[d4 + 1] + w * h.y;
    o.z = bias[d4 + 2] + w * h.z;
    o.w = bias[d4 + 3] + w * h.w;
    *reinterpret_cast<float4*>(out + (size_t)i * D + d4) = o;
}

// ---------------- edge scatter: out[dst] += dinv[src]*dinv[dst] * h[src] ----
__global__ void gcn_scatter(const int* __restrict__ src,
                            const int* __restrict__ dst,
                            const float* __restrict__ dinv,
                            const float* __restrict__ H,
                            float* __restrict__ out, int e, int D) {
    int tid = blockIdx.x * blockDim.x + threadIdx.x;
    int vecPerRow = D >> 2;
    int ei = tid / vecPerRow;
    int d4 = (tid - ei * vecPerRow) * 4;
    if (ei >= e) return;
    int s = src[ei];
    int t = dst[ei];
    float norm = dinv[s] * dinv[t];
    float4 h = *reinterpret_cast<const float4*>(H + (size_t)s * D + d4);
    float* op = out + (size_t)t * D + d4;
    atomicAdd(op + 0, norm * h.x);
    atomicAdd(op + 1, norm * h.y);
    atomicAdd(op + 2, norm * h.z);
    atomicAdd(op + 3, norm * h.w);
}

// ---------------- log_softmax over 32-wide rows (one wave per row) ----------
__global__ void gcn_log_softmax32(float* __restrict__ out, int n) {
    int wave = (blockIdx.x * blockDim.x + threadIdx.x) >> 5;
    int lane = threadIdx.x & 31;
    if (wave >= n) return;
    float x = out[(size_t)wave * 32 + lane];
    float m = x;
    #pragma unroll
    for (int off = 16; off > 0; off >>= 1)
        m = fmaxf(m, __shfl_xor(m, off, 32));
    float ex = expf(x - m);
    float s = ex;
    #pragma unroll
    for (int off = 16; off > 0; off >>= 1)
        s += __shfl_xor(s, off, 32);
    out[(size_t)wave * 32 + lane] = x - m - logf(s);
}

// ---------------------------------------------------------------------------

extern "C" void kernel_launch(void* const* d_in, const int* in_sizes, int n_in,
                              void* d_out, int out_size, void* d_ws, size_t ws_size,
                              hipStream_t stream) {
    const int*   edges = (const int*)  d_in[0];   // [2, E] int32
    const float* feat  = (const float*)d_in[1];   // [N, d0]
    const float* W0 = (const float*)d_in[2];
    const float* b0 = (const float*)d_in[3];
    const float* W1 = (const float*)d_in[4];
    const float* b1 = (const float*)d_in[5];
    const float* W2 = (const float*)d_in[6];
    const float* b2 = (const float*)d_in[7];

    const int E  = in_sizes[0] / 2;
    const int d1 = in_sizes[3];          // 64
    const int d2 = in_sizes[5];          // 64
    const int d3 = in_sizes[7];          // 32
    const int d0 = in_sizes[2] / d1;     // 64
    const int N  = in_sizes[1] / d0;     // 100000 (divisible by 16)

    const int* srcIdx = edges;           // edges[0]
    const int* dstIdx = edges + E;       // edges[1]

    float* dinv = (float*)d_ws;          // N
    float* bufA = dinv + N;              // N * 64  (transform output h)
    float* bufB = bufA + (size_t)N * 64; // N * 64  (aggregation output)
    float* outF = (float*)d_out;         // N * d3

    const int TB = 256;
    auto cdiv = [](long long a, long long b) { return (int)((a + b - 1) / b); };

    const bool fast = (d0 == 64 && d1 == 64 && d2 == 64 && d3 == 32 && (N & 15) == 0);

    // --- normalization coefficients ---
    gcn_deg_init <<<cdiv(N, TB), TB, 0, stream>>>(dinv, N);
    gcn_deg_count<<<cdiv(E, TB), TB, 0, stream>>>(dstIdx, dinv, E);
    gcn_deg_rsqrt<<<cdiv(N, TB), TB, 0, stream>>>(dinv, N);

    // --- layer 0: h = feat @ W0 ; agg -> bufB ---
    if (fast) {
        long long waves = (long long)(N / 16) * (64 / 32);
        gcn_gemm_wmma_t<64, 64, false><<<cdiv(waves * 32, TB), TB, 0, stream>>>(feat, W0, bufA, N);
    } else {
        long long waves = (long long)cdiv(N, 16) * (d1 / 16);
        gcn_gemm_wmma_g<<<cdiv(waves * 32, TB), TB, 0, stream>>>(feat, W0, bufA, N, d0, d1, 0);
    }
    gcn_agg_init<<<cdiv((long long)N * (d1 / 4), TB), TB, 0, stream>>>(bufA, dinv, b0, bufB, N, d1);
    gcn_scatter <<<cdiv((long long)E * (d1 / 4), TB), TB, 0, stream>>>(srcIdx, dstIdx, dinv, bufA, bufB, E, d1);

    // --- layer 1: h = relu(bufB) @ W1 ; agg -> bufB ---
    if (fast) {
        long long waves = (long long)(N / 16) * (64 / 32);
        gcn_gemm_wmma_t<64, 64, true><<<cdiv(waves * 32, TB), TB, 0, stream>>>(bufB, W1, bufA, N);
    } else {
        long long waves = (long long)cdiv(N, 16) * (d2 / 16);
        gcn_gemm_wmma_g<<<cdiv(waves * 32, TB), TB, 0, stream>>>(bufB, W1, bufA, N, d1, d2, 1);
    }
    gcn_agg_init<<<cdiv((long long)N * (d2 / 4), TB), TB, 0, stream>>>(bufA, dinv, b1, bufB, N, d2);
    gcn_scatter <<<cdiv((long long)E * (d2 / 4), TB), TB, 0, stream>>>(srcIdx, dstIdx, dinv, bufA, bufB, E, d2);

    // --- layer 2: h = relu(bufB) @ W2 ; agg -> d_out ; log_softmax in place ---
    if (fast) {
        long long waves = (long long)(N / 16) * (32 / 32);
        gcn_gemm_wmma_t<64, 32, true><<<cdiv(waves * 32, TB), TB, 0, stream>>>(bufB, W2, bufA, N);
    } else {
        long long waves = (long long)cdiv(N, 16) * (d3 / 16);
        gcn_gemm_wmma_g<<<cdiv(waves * 32, TB), TB, 0, stream>>>(bufB, W2, bufA, N, d2, d3, 1);
    }
    gcn_agg_init<<<cdiv((long long)N * (d3 / 4), TB), TB, 0, stream>>>(bufA, dinv, b2, outF, N, d3);
    gcn_scatter <<<cdiv((long long)E * (d3 / 4), TB), TB, 0, stream>>>(srcIdx, dstIdx, dinv, bufA, outF, E, d3);
    if (d3 == 32) {
        gcn_log_softmax32<<<cdiv((long long)N * 32, TB), TB, 0, stream>>>(outF, N);
    }
}